// Seq2Seq_30356828848453
// MI455X (gfx1250) — compile-verified
//
#include <hip/hip_runtime.h>
#include <hip/hip_bf16.h>

// ---------------------------------------------------------------------------
// CVAE Seq2Seq (encoder LSTM -> latent -> decoder LSTM -> vocab projection)
// fp32 end-to-end. Matrix work: V_WMMA_F32_16X16X4_F32.
// LDS staging: GLOBAL_LOAD_ASYNC_TO_LDS_B128 + s_wait_asynccnt (double-buffered).
// ---------------------------------------------------------------------------

typedef __attribute__((ext_vector_type(2))) float v2f;
typedef __attribute__((ext_vector_type(8))) float v8f;

#define HID   1024
#define G4H   4096
#define SEQT  256
#define LAT   256
#define CND   64
#define VOC   32000
#define LSTM_WGS 32

// GEMM tiling
#define MT    64            // M rows per block
#define NT    128           // N cols per block (8 waves x 16)
#define KC    128           // K chunk staged in LDS
#define LDST  (KC + 4)      // padded LDS row stride (floats) -> conflict-free frags
#define BUFSZ (MT * LDST)   // floats per A buffer

__device__ __forceinline__ float sigmoidf_(float x) {
    return 1.0f / (1.0f + expf(-x));
}

// ---- CDNA5 async global->LDS copy (ASYNCcnt-tracked) ----------------------
// NOTE: lds_base is passed (unused in the template) so the shared array's
// address ESCAPES into an instruction with a memory clobber; otherwise LLVM
// proves "no stores to this LDS global" and folds all its loads to undef.
__device__ __forceinline__ void async_load_b128(unsigned int lds_byte_off,
                                                const void* gaddr,
                                                void* lds_base) {
    asm volatile("global_load_async_to_lds_b128 %0, %1, off"
                 :: "v"(lds_byte_off), "v"(gaddr), "v"(lds_base)
                 : "memory");
}
__device__ __forceinline__ void wait_asynccnt_0() {
    asm volatile("s_wait_asynccnt 0x0" ::: "memory");
}
__device__ __forceinline__ void wait_asynccnt_8() {
    asm volatile("s_wait_asynccnt 0x8" ::: "memory");
}

// Device-wide barrier: monotonic counter, one arrival per workgroup.
// Acquire-poll by lane 0 of EVERY wave so each wave's cache path sees the
// invalidate. s_cluster_barrier is a hardware NOP when not cluster-launched.
__device__ __forceinline__ void grid_barrier(unsigned int* ctr, unsigned int target) {
    __syncthreads();
    if (threadIdx.x == 0) {
        __builtin_amdgcn_s_cluster_barrier();
        __threadfence();
        __hip_atomic_fetch_add(ctr, 1u, __ATOMIC_RELEASE, __HIP_MEMORY_SCOPE_AGENT);
    }
    if ((threadIdx.x & 31) == 0) {
        while (__hip_atomic_load(ctr, __ATOMIC_ACQUIRE, __HIP_MEMORY_SCOPE_AGENT) < target) {
            __builtin_amdgcn_s_sleep(8);
        }
    }
    __syncthreads();
}

// ---------------------------------------------------------------------------
// C[M,N] = A[M,1024] @ W[N,1024]^T + bias[N]
// Block = 256 threads (8 waves). Tile: 64(M) x 128(N); wave owns 16 N cols and
// 4 M-subtiles, reusing each B fragment for 4 WMMAs. A staged in LDS by
// double-buffered async B128 copies, K chunked by 128.
//
// Fragment layout (ISA 7.12.2, 32-bit):
//   lane l: half=l>>4, l16=l&15
//   A frag: a = A[l16][k0 + 2*half .. +1]
//   B frag: b = W^T[k0 + 2*half .. +1][n] = W[n][k0+2*half .. +1]
//   C frag: acc[r] -> C[r + 8*half][l16]
// ---------------------------------------------------------------------------
__global__ __launch_bounds__(256) void k_gemm_f32wmma(
    const float* __restrict__ A,
    const float* __restrict__ W,
    const float* __restrict__ bias,
    float* __restrict__ C,
    int N)
{
    __shared__ float Abuf[2 * BUFSZ];   // ONLY shared object -> LDS offset 0

    const int m0  = blockIdx.y * MT;
    const int n0b = blockIdx.x * NT;
    const int tid = threadIdx.x;
    const int wave = tid >> 5;
    const int lane = tid & 31;
    const int half = lane >> 4;
    const int l16  = lane & 15;
    const int n0   = n0b + wave * 16;

    const float* wrow = &W[(size_t)(n0 + l16) * HID + 2 * half];

    // issue async stage of a 64x128 A chunk into buffer `buf`
    auto stage = [&](int kc, int buf) {
#pragma unroll
        for (int rnd = 0; rnd < 8; ++rnd) {
            const int idx = rnd * 1024 + tid * 4;        // 64*128 floats total
            const int r   = idx >> 7;                    // /128
            const int c   = idx & (KC - 1);
            const unsigned int lds_off =
                (unsigned int)((buf * BUFSZ + r * LDST + c) * 4);
            async_load_b128(lds_off, &A[(size_t)(m0 + r) * HID + kc + c], Abuf);
        }
    };

    v8f acc[4] = {v8f{}, v8f{}, v8f{}, v8f{}};

    stage(0, 0);
    const int NCHUNK = HID / KC;                         // 8
    for (int ch = 0; ch < NCHUNK; ++ch) {
        if (ch + 1 < NCHUNK) {
            stage((ch + 1) * KC, (ch + 1) & 1);          // overlap next chunk
            wait_asynccnt_8();                           // current chunk landed
        } else {
            wait_asynccnt_0();
        }
        __syncthreads();

        const float* abase = &Abuf[(ch & 1) * BUFSZ + l16 * LDST + 2 * half];
        const float* wk    = wrow + ch * KC;
        // prefetch next W chunk (4 cachelines/lane)
        if (ch + 1 < NCHUNK) {
#pragma unroll
            for (int pf = 0; pf < 4; ++pf)
                __builtin_prefetch(wk + KC + pf * 32, 0, 1);
        }
#pragma unroll 4
        for (int k0 = 0; k0 < KC; k0 += 4) {
            const v2f b  = *(const v2f*)(wk + k0);
            const v2f a0 = *(const v2f*)(abase + 0 * 16 * LDST + k0);
            const v2f a1 = *(const v2f*)(abase + 1 * 16 * LDST + k0);
            const v2f a2 = *(const v2f*)(abase + 2 * 16 * LDST + k0);
            const v2f a3 = *(const v2f*)(abase + 3 * 16 * LDST + k0);
            acc[0] = __builtin_amdgcn_wmma_f32_16x16x4_f32(false, a0, false, b, (short)0, acc[0], false, false);
            acc[1] = __builtin_amdgcn_wmma_f32_16x16x4_f32(false, a1, false, b, (short)0, acc[1], false, false);
            acc[2] = __builtin_amdgcn_wmma_f32_16x16x4_f32(false, a2, false, b, (short)0, acc[2], false, false);
            acc[3] = __builtin_amdgcn_wmma_f32_16x16x4_f32(false, a3, false, b, (short)0, acc[3], false, false);
        }
        __syncthreads();                                 // buffer reuse safety
    }

    const int n = n0 + l16;
    const float bn = bias[n];
#pragma unroll
    for (int sub = 0; sub < 4; ++sub) {
#pragma unroll
        for (int r = 0; r < 8; ++r) {
            const int m = m0 + sub * 16 + r + 8 * half;
            C[(size_t)m * N + n] = acc[sub][r] + bn;
        }
    }
}

// ---------------------------------------------------------------------------
// Persistent LSTM recurrence. Gx = X@Wih^T + bih precomputed (WMMA GEMM).
// 32 workgroups; WG w owns hidden slice j0=w*32..+31 and gate rows
// {j, j+1024, j+2048, j+3072} so the cell update is WG-local.
// One device-wide barrier per timestep; h broadcast staged via async B128.
// ---------------------------------------------------------------------------
__global__ __launch_bounds__(256) void k_lstm(
    const float* __restrict__ Gx,    // (T, 4096)
    const float* __restrict__ whh,   // (4096, 1024)
    const float* __restrict__ bhh,   // (4096)
    const float* __restrict__ h0,    // (1024)
    const float* __restrict__ c0,    // (1024)
    float* __restrict__ hbuf,        // (1024) device-shared h scratch
    float* __restrict__ hs,          // (T, 1024)
    unsigned int* __restrict__ bar,
    int T)
{
    __shared__ float smem[HID + 128 + 32];  // [h(1024) | g(128) | c(32)], h at LDS off 0
    float* h_lds = smem;
    float* g_lds = smem + HID;
    float* c_lds = smem + HID + 128;

    const int wg  = blockIdx.x;      // 0..31
    const int tid = threadIdx.x;
    const int j0  = wg * 32;

    if (tid < 32) {
        c_lds[tid]     = c0[j0 + tid];
        hbuf[j0 + tid] = h0[j0 + tid];
    }
    grid_barrier(bar, LSTM_WGS * 1u);

    const int wave = tid >> 5;
    const int lane = tid & 31;

    for (int t = 0; t < T; ++t) {
        // stage full h vector: one async B128 per thread
        async_load_b128((unsigned int)(tid * 16), &hbuf[tid * 4], smem);
        wait_asynccnt_0();
        __syncthreads();

        // 16 gate rows per wave, wave-cooperative dot products
        for (int rr = 0; rr < 16; ++rr) {
            const int lr   = wave * 16 + rr;        // 0..127
            const int gate = lr >> 5;               // 0..3 (i,f,g,o)
            const int jj   = lr & 31;
            const int r    = gate * HID + j0 + jj;
            const float* wr = &whh[(size_t)r * HID];
            float p = 0.f;
#pragma unroll 8
            for (int k = lane; k < HID; k += 32)
                p = fmaf(h_lds[k], wr[k], p);
#pragma unroll
            for (int off = 16; off; off >>= 1)
                p += __shfl_down(p, off, 32);
            if (lane == 0)
                g_lds[lr] = p + Gx[(size_t)t * G4H + r] + bhh[r];
        }
        __syncthreads();

        if (tid < 32) {
            const float gi = g_lds[tid];
            const float gf = g_lds[32 + tid];
            const float gg = g_lds[64 + tid];
            const float go = g_lds[96 + tid];
            float c = sigmoidf_(gf) * c_lds[tid] + sigmoidf_(gi) * tanhf(gg);
            float h = sigmoidf_(go) * tanhf(c);
            c_lds[tid] = c;
            hbuf[j0 + tid] = h;
            hs[(size_t)t * HID + j0 + tid] = h;
        }
        grid_barrier(bar, LSTM_WGS * (unsigned)(t + 2));
    }
}

// ---------------------------------------------------------------------------
// Small kernels
// ---------------------------------------------------------------------------
__global__ void k_init(unsigned int* bar) {
    if (threadIdx.x < 64) bar[threadIdx.x] = 0u;
}

__global__ __launch_bounds__(256) void k_gather(
    const float* __restrict__ emb, const int* __restrict__ ids,
    float* __restrict__ X, int shift)
{
    const int t  = blockIdx.x;
    const int id = shift ? (t == 0 ? 0 : ids[t - 1]) : ids[t];
    const float4* src = (const float4*)&emb[(size_t)id * HID];
    float4* dst = (float4*)&X[(size_t)t * HID];
    dst[threadIdx.x] = src[threadIdx.x];
}

__global__ void k_prep(const float* __restrict__ cond_emb,
                       const int* __restrict__ input_c,
                       float* __restrict__ h0, float* __restrict__ c0)
{
    const int i = blockIdx.x * blockDim.x + threadIdx.x;
    if (i < HID) {
        c0[i] = 0.f;
        h0[i] = (i < HID - CND) ? 0.f
                                : cond_emb[input_c[0] * CND + (i - (HID - CND))];
    }
}

__global__ __launch_bounds__(256) void k_latent(
    const float* __restrict__ hT,
    const float* __restrict__ mean_w, const float* __restrict__ mean_b,
    const float* __restrict__ lgv_w,  const float* __restrict__ lgv_b,
    const float* __restrict__ eps,
    const float* __restrict__ cond_emb, const int* __restrict__ target_c,
    float* __restrict__ m_out, float* __restrict__ lgv_out,
    float* __restrict__ zc)
{
    __shared__ float h_s[HID];
    const int tid = threadIdx.x;
    for (int i = tid; i < HID; i += 256) h_s[i] = hT[i];
    __syncthreads();

    const float* mr = &mean_w[(size_t)tid * HID];
    const float* lr = &lgv_w[(size_t)tid * HID];
    float sm = 0.f, sl = 0.f;
    for (int k = 0; k < HID; ++k) {
        sm = fmaf(h_s[k], mr[k], sm);
        sl = fmaf(h_s[k], lr[k], sl);
    }
    const float m  = sm + mean_b[tid];
    const float lg = sl + lgv_b[tid];
    m_out[tid]   = m;
    lgv_out[tid] = lg;
    zc[tid]      = eps[tid] * expf(lg * 0.5f) + m;
    if (tid < CND) zc[LAT + tid] = cond_emb[target_c[0] * CND + tid];
}

__global__ void k_l2hc(const float* __restrict__ l2h_w, const float* __restrict__ l2h_b,
                       const float* __restrict__ l2c_w, const float* __restrict__ l2c_b,
                       const float* __restrict__ zc,
                       float* __restrict__ dh0, float* __restrict__ dc0)
{
    const int i = blockIdx.x * blockDim.x + threadIdx.x;   // 0..2047
    if (i >= 2 * HID) return;
    const int row = i & (HID - 1);
    const float* Wm = (i < HID) ? l2h_w : l2c_w;
    const float* bb = (i < HID) ? l2h_b : l2c_b;
    const float* wr = &Wm[(size_t)row * (LAT + CND)];
    float s = 0.f;
    for (int k = 0; k < LAT + CND; ++k) s = fmaf(zc[k], wr[k], s);
    float r = s + bb[row];
    if (i < HID) dh0[row] = r; else dc0[row] = r;
}

// ---------------------------------------------------------------------------
extern "C" void kernel_launch(void* const* d_in, const int* in_sizes, int n_in,
                              void* d_out, int out_size, void* d_ws, size_t ws_size,
                              hipStream_t stream)
{
    (void)in_sizes; (void)n_in; (void)out_size; (void)ws_size;

    const float* cond_emb = (const float*)d_in[0];
    const float* enc_emb  = (const float*)d_in[1];
    const float* enc_wih  = (const float*)d_in[2];
    const float* enc_whh  = (const float*)d_in[3];
    const float* enc_bih  = (const float*)d_in[4];
    const float* enc_bhh  = (const float*)d_in[5];
    const float* mean_w   = (const float*)d_in[6];
    const float* mean_b   = (const float*)d_in[7];
    const float* lgv_w    = (const float*)d_in[8];
    const float* lgv_b    = (const float*)d_in[9];
    const float* l2h_w    = (const float*)d_in[10];
    const float* l2h_b    = (const float*)d_in[11];
    const float* l2c_w    = (const float*)d_in[12];
    const float* l2c_b    = (const float*)d_in[13];
    const float* dec_emb  = (const float*)d_in[14];
    const float* dec_wih  = (const float*)d_in[15];
    const float* dec_whh  = (const float*)d_in[16];
    const float* dec_bih  = (const float*)d_in[17];
    const float* dec_bhh  = (const float*)d_in[18];
    const float* out_w    = (const float*)d_in[19];
    const float* out_b    = (const float*)d_in[20];
    const float* eps      = (const float*)d_in[21];
    const int*   input_ids  = (const int*)d_in[22];
    const int*   target_ids = (const int*)d_in[23];
    const int*   input_c    = (const int*)d_in[24];
    const int*   target_c   = (const int*)d_in[25];

    float* ws = (float*)d_ws;
    size_t off = 0;
    float* enc_x  = ws + off; off += (size_t)SEQT * HID;
    float* dec_x  = ws + off; off += (size_t)SEQT * HID;
    float* gx_enc = ws + off; off += (size_t)SEQT * G4H;
    float* gx_dec = ws + off; off += (size_t)SEQT * G4H;
    float* hs_enc = ws + off; off += (size_t)SEQT * HID;
    float* hs_dec = ws + off; off += (size_t)SEQT * HID;
    float* h0e    = ws + off; off += HID;
    float* c0e    = ws + off; off += HID;
    float* dh0    = ws + off; off += HID;
    float* dc0    = ws + off; off += HID;
    float* hbufe  = ws + off; off += HID;
    float* hbufd  = ws + off; off += HID;
    float* zc     = ws + off; off += 512;
    unsigned int* bar = (unsigned int*)(ws + off);

    float* logits  = (float*)d_out;                       // (256, 32000)
    float* m_out   = logits + (size_t)SEQT * VOC;         // (256)
    float* lgv_out = m_out + LAT;                         // (256)

    k_init<<<1, 64, 0, stream>>>(bar);
    k_gather<<<SEQT, 256, 0, stream>>>(enc_emb, input_ids, enc_x, 0);
    k_gather<<<SEQT, 256, 0, stream>>>(dec_emb, target_ids, dec_x, 1);
    k_prep<<<4, 256, 0, stream>>>(cond_emb, input_c, h0e, c0e);

    dim3 gIn(G4H / NT, SEQT / MT);                        // (32, 4)
    k_gemm_f32wmma<<<gIn, 256, 0, stream>>>(enc_x, enc_wih, enc_bih, gx_enc, G4H);
    k_gemm_f32wmma<<<gIn, 256, 0, stream>>>(dec_x, dec_wih, dec_bih, gx_dec, G4H);

    k_lstm<<<LSTM_WGS, 256, 0, stream>>>(gx_enc, enc_whh, enc_bhh, h0e, c0e,
                                         hbufe, hs_enc, bar + 0, SEQT);

    k_latent<<<1, 256, 0, stream>>>(hs_enc + (size_t)(SEQT - 1) * HID,
                                    mean_w, mean_b, lgv_w, lgv_b, eps,
                                    cond_emb, target_c, m_out, lgv_out, zc);
    k_l2hc<<<8, 256, 0, stream>>>(l2h_w, l2h_b, l2c_w, l2c_b, zc, dh0, dc0);

    k_lstm<<<LSTM_WGS, 256, 0, stream>>>(gx_dec, dec_whh, dec_bhh, dh0, dc0,
                                         hbufd, hs_dec, bar + 32, SEQT);

    dim3 gOut(VOC / NT, SEQT / MT);                       // (250, 4)
    k_gemm_f32wmma<<<gOut, 256, 0, stream>>>(hs_dec, out_w, out_b, logits, VOC);
}